// RD_loss8_16561393893906
// MI455X (gfx1250) — compile-verified
//
#include <hip/hip_runtime.h>

#define RD_B    16
#define RD_LOGN 18
#define RD_N    (1 << RD_LOGN)       // 262144 = C*H*W
#define RD_R    5
#define RD_DEN  20971520.0           // B * N * R

typedef float v2f __attribute__((ext_vector_type(2)));
typedef float v8f __attribute__((ext_vector_type(8)));

// Full-wave (32-lane) sum. Uses V_WMMA_F32_16X16X4_F32 with A = ones(16x4),
// B packed with each lane's value (b0 = v, b1 = 0). The 2 A/B VGPRs x 32
// lanes map bijectively onto the 64 matrix entries, so D[m,n] = sum_k B[k,n]
// and sum over row M=0 (D vgpr0, lanes 0..15) == sum of all lane values,
// independent of the exact in-matrix slot mapping.
__device__ __forceinline__ float wave_sum32(float v) {
#if defined(__gfx1250__) && __has_builtin(__builtin_amdgcn_wmma_f32_16x16x4_f32)
    v2f a; a[0] = 1.0f; a[1] = 1.0f;
    v2f b; b[0] = v;    b[1] = 0.0f;
    v8f c = {};
    v8f d = __builtin_amdgcn_wmma_f32_16x16x4_f32(
        /*neg_a=*/false, a, /*neg_b=*/false, b,
        /*c_mod=*/(short)0, c, /*reuse_a=*/false, /*reuse_b=*/false);
    float s = d[0];                 // lanes 0..15: D[0, n]; lanes 16..31: D[8, n]
#else
    float s = v;
    s += __shfl_xor(s, 16, 32);
#endif
    // reduce within each 16-lane half: lane 0 (and lane 16) hold the total
    s += __shfl_xor(s, 1, 32);
    s += __shfl_xor(s, 2, 32);
    s += __shfl_xor(s, 4, 32);
    s += __shfl_xor(s, 8, 32);
    return s;                       // valid on lanes where (lane & 15) == 0
}

__global__ void rd_init_kernel(double* __restrict__ acc) { acc[0] = 0.0; }

__global__ void __launch_bounds__(256) rd_main_kernel(
    const float* __restrict__ pred,
    const float* __restrict__ gt,
    const int*   __restrict__ perms,
    double*      __restrict__ acc)
{
    __shared__ float s_sum;
    if (threadIdx.x == 0) s_sum = 0.0f;
    __syncthreads();

    const int nvec   = (RD_B << RD_LOGN) >> 2;                 // 1,048,576 float4 units
    const int tid    = blockIdx.x * blockDim.x + threadIdx.x;
    const int stride = gridDim.x * blockDim.x;                 // sized to divide nvec evenly

    float acc_l = 0.0f;

    for (int u = tid; u < nvec; u += stride) {
        const int e = u << 2;                                  // element index (same batch for all 4)
        const int b = e >> RD_LOGN;
        const int i = e & (RD_N - 1);
        const float* pb = pred + ((size_t)b << RD_LOGN);
        const float* gb = gt   + ((size_t)b << RD_LOGN);

        // prefetch next iteration's streamed lines (speculative; dropped if OOB)
        if (u + stride < nvec) {
            const int e2 = (u + stride) << 2;
            __builtin_prefetch(pred  + e2, 0, 3);
            __builtin_prefetch(gt    + e2, 0, 3);
            __builtin_prefetch(perms + e2, 0, 3);
        }

        const float4 p4 = *(const float4*)(pb + i);
        const float4 g4 = *(const float4*)(gb + i);

        int4 idx[RD_R];
        #pragma unroll
        for (int r = 0; r < RD_R; ++r)
            idx[r] = *(const int4*)(perms + (((size_t)(r * RD_B + b)) << RD_LOGN) + i);

        #pragma unroll
        for (int j = 0; j < 4; ++j) {
            const float pj = (&p4.x)[j];
            const float gj = (&g4.x)[j];
            float pd[RD_R], gd[RD_R];
            float sp = 0.0f, sg = 0.0f;
            #pragma unroll
            for (int r = 0; r < RD_R; ++r) {
                const int p = (&idx[r].x)[j];                  // gather index within batch slice
                const float dv = pj - pb[p];                   // L2-resident random gather
                const float ev = gj - gb[p];
                pd[r] = dv; gd[r] = ev;
                sp += dv * dv; sg += ev * ev;
            }
            float np = sqrtf(sp); np = (np == 0.0f) ? 1.0f : np;
            float ng = sqrtf(sg); ng = (ng == 0.0f) ? 1.0f : ng;
            const float ip = 1.0f / np;
            const float ig = 1.0f / ng;
            #pragma unroll
            for (int r = 0; r < RD_R; ++r)
                acc_l += fabsf(pd[r] * ip - gd[r] * ig);
        }
    }

    // EXEC is all-ones here (uniform trip count, full blocks) -> WMMA legal.
    const float wsum = wave_sum32(acc_l);

    if ((threadIdx.x & 31) == 0)
        atomicAdd(&s_sum, wsum);                               // ds_add_f32
    __syncthreads();

    if (threadIdx.x == 0)
        atomicAdd(acc, (double)s_sum);                         // global f64 atomic
}

__global__ void rd_final_kernel(const double* __restrict__ acc, float* __restrict__ out) {
    out[0] = (float)(acc[0] / RD_DEN);
}

extern "C" void kernel_launch(void* const* d_in, const int* in_sizes, int n_in,
                              void* d_out, int out_size, void* d_ws, size_t ws_size,
                              hipStream_t stream) {
    const float* pred  = (const float*)d_in[0];
    const float* gt    = (const float*)d_in[1];
    const int*   perms = (const int*)d_in[2];
    double* acc = (double*)d_ws;
    float*  out = (float*)d_out;

    rd_init_kernel<<<1, 1, 0, stream>>>(acc);
    // 2048 blocks x 256 threads = 524288 threads; 1048576 float4 units -> exactly
    // 2 iterations per thread (no divergence, EXEC stays all-ones for WMMA).
    rd_main_kernel<<<2048, 256, 0, stream>>>(pred, gt, perms, acc);
    rd_final_kernel<<<1, 1, 0, stream>>>(acc, out);
}